// AbstractGCN_4123168604344
// MI455X (gfx1250) — compile-verified
//
#include <hip/hip_runtime.h>
#include <hip/hip_bf16.h>

// ---------------------------------------------------------------------------
// AbstractGCN on MI455X (gfx1250, wave32).
//   out = tanh( x + sum_s spmm_s( (tanh(spmm_s(x@W0^T+b0)))@W1^T+b1 ) )
// - Dense GEMMs: V_WMMA_F32_16X16X4_F32 (fp32 matrix pipe; GEMM is
//   memory-bound at AI~32 flop/byte vs 23.3 TB/s HBM, so fp32 is free).
// - SpMM: 51.2MB feature matrix is L2-resident (192MB L2). Wave owns 32
//   edges: coalesced edge-list load + v_readlane broadcast, float4/lane
//   gather, inline-asm global_atomic_add_f32 scatter.
// ---------------------------------------------------------------------------

#define NNODES 100000
#define DIM    128
#define NEDGES 3200000
#define BR     64      // GEMM rows per block

typedef __attribute__((ext_vector_type(2))) float v2f;
typedef __attribute__((ext_vector_type(8))) float v8f;

__device__ __forceinline__ void atomic_add_f32(float* p, float v) {
    // no-return HW float atomic; tracked by STOREcnt, s_endpgm waits it out
    asm volatile("global_atomic_add_f32 %0, %1, off" :: "v"(p), "v"(v) : "memory");
}

// ---------------------------------------------------------------------------
// Dense GEMM: out[r, o] = sum_k A[r,k] * W[o,k] + bias[o]
//   (optionally A := tanh(A) on load, fused into LDS staging)
// Block: 64 rows x 128 cols, 256 threads = 8 waves.
// Wave w: rows 16*(w>>1).. , cols 64*(w&1).. -> 4 accumulators of 16x16.
// K-loop: 32 steps of V_WMMA_F32_16X16X4_F32.
// ---------------------------------------------------------------------------
__global__ __launch_bounds__(256) void gcn_gemm(const float* __restrict__ A,
                                                const float* __restrict__ W,
                                                const float* __restrict__ bias,
                                                float* __restrict__ out,
                                                int nrows, int applyTanh)
{
    __shared__ float Atile[BR * DIM];      // 32 KB
    __shared__ float WtP[DIM * DIM];       // 64 KB: WtP[(k>>1)*256 + o*2 + (k&1)] = W[o*128+k]

    const int tid = threadIdx.x;
    const int rowBase = blockIdx.x * BR;

    // stage weights: float4 global reads, paired b64 LDS writes (transposed)
    #pragma unroll
    for (int i = 0; i < (DIM * DIM) / (256 * 4); ++i) {   // 16 iters
        int idx4 = i * 256 + tid;
        int o  = idx4 >> 5;              // 32 float4 per 128-float row
        int kb = (idx4 & 31) * 4;
        float4 v = *(const float4*)(W + (size_t)o * DIM + kb);
        *(v2f*)(WtP + (kb >> 1) * 256 + o * 2)       = (v2f){v.x, v.y};
        *(v2f*)(WtP + ((kb + 2) >> 1) * 256 + o * 2) = (v2f){v.z, v.w};
    }
    // stage A tile (coalesced float4), optional fused tanh, zero-pad past nrows
    #pragma unroll
    for (int i = 0; i < (BR * DIM) / (256 * 4); ++i) {    // 8 iters
        int idx = (i * 256 + tid) * 4;
        int r = rowBase + (idx >> 7);
        float4 v = make_float4(0.f, 0.f, 0.f, 0.f);
        if (r < nrows) v = *(const float4*)(A + (size_t)r * DIM + (idx & 127));
        if (applyTanh) {
            v.x = tanhf(v.x); v.y = tanhf(v.y); v.z = tanhf(v.z); v.w = tanhf(v.w);
        }
        *(float4*)(Atile + idx) = v;
    }
    __syncthreads();

    const int wid     = tid >> 5;
    const int lane    = tid & 31;
    const int waveRow = (wid >> 1) * 16;   // 0,16,32,48
    const int waveCol = (wid & 1) * 64;    // 0 or 64
    const int m       = lane & 15;         // row (A) / col (B,C) within tile
    const int kk      = (lane >> 4) * 2;   // K sub-offset for lanes 16..31

    // init accumulators with bias (bias[o] replicated across the 8 M-rows/lane)
    v8f acc[4];
    #pragma unroll
    for (int t = 0; t < 4; ++t) {
        float b = bias[waveCol + 16 * t + m];
        acc[t] = (v8f){b, b, b, b, b, b, b, b};
    }

    #pragma unroll 4
    for (int k0 = 0; k0 < DIM; k0 += 4) {
        // A frag 16x4 f32: lane<16 -> (K=k0,k0+1), lane>=16 -> (K=k0+2,k0+3)
        v2f a = *(const v2f*)(Atile + (waveRow + m) * DIM + k0 + kk);
        int p = (k0 + kk) >> 1;
        #pragma unroll
        for (int t = 0; t < 4; ++t) {
            v2f b = *(const v2f*)(WtP + p * 256 + (waveCol + 16 * t + m) * 2);
            acc[t] = __builtin_amdgcn_wmma_f32_16x16x4_f32(
                false, a, false, b, (short)0, acc[t], false, false);
        }
    }

    // C/D layout: VGPR v, lanes 0-15 -> M=v, lanes 16-31 -> M=v+8; N = lane&15
    const int mhi = (lane >> 4) * 8;
    if (rowBase + BR <= nrows) {
        // fast path: 1562 of 1563 blocks, branch-free stores
        #pragma unroll
        for (int t = 0; t < 4; ++t) {
            int col = waveCol + 16 * t + m;
            #pragma unroll
            for (int v = 0; v < 8; ++v) {
                int r = rowBase + waveRow + v + mhi;
                out[(size_t)r * DIM + col] = acc[t][v];
            }
        }
    } else {
        #pragma unroll
        for (int t = 0; t < 4; ++t) {
            int col = waveCol + 16 * t + m;
            #pragma unroll
            for (int v = 0; v < 8; ++v) {
                int r = rowBase + waveRow + v + mhi;
                if (r < nrows) out[(size_t)r * DIM + col] = acc[t][v];
            }
        }
    }
}

// ---------------------------------------------------------------------------
// COO SpMM: y[rows[e], :] += vals[e] * h[cols[e], :]
// One wave32 per 32 edges: coalesced edge-list loads, v_readlane broadcast,
// lane L handles features [4L, 4L+4) of each edge's row.
// ---------------------------------------------------------------------------
__global__ __launch_bounds__(256) void gcn_spmm(const float* __restrict__ vals,
                                                const int*   __restrict__ rows,
                                                const int*   __restrict__ cols,
                                                const float* __restrict__ h,
                                                float*       __restrict__ y,
                                                int nEdges)
{
    const int lane = threadIdx.x & 31;
    const int wave = (int)((blockIdx.x * 256u + threadIdx.x) >> 5);
    const int e0   = wave * 32;
    if (e0 >= nEdges) return;

    // one coalesced load of 32 edges' metadata (NEDGES % 32 == 0)
    int   rv = rows[e0 + lane];
    int   cv = cols[e0 + lane];
    float vv = vals[e0 + lane];

    #pragma unroll
    for (int i = 0; i < 32; ++i) {
        int   r = __builtin_amdgcn_readlane(rv, i);                      // SGPR
        int   c = __builtin_amdgcn_readlane(cv, i);                      // SGPR
        float v = __int_as_float(
                      __builtin_amdgcn_readlane(__float_as_int(vv), i)); // SGPR

        float4 d = ((const float4*)(h + (size_t)c * DIM))[lane];  // 512B/wave
        float* dst = y + (size_t)r * DIM + lane * 4;
        atomic_add_f32(dst + 0, v * d.x);
        atomic_add_f32(dst + 1, v * d.y);
        atomic_add_f32(dst + 2, v * d.z);
        atomic_add_f32(dst + 3, v * d.w);
    }
}

// ---------------------------------------------------------------------------
__global__ __launch_bounds__(256) void gcn_zero(float4* __restrict__ p, int n4)
{
    int i = blockIdx.x * 256 + threadIdx.x;
    if (i < n4) p[i] = make_float4(0.f, 0.f, 0.f, 0.f);
}

// out = tanh(out + x), in place on d_out (which accumulated h_sum)
__global__ __launch_bounds__(256) void gcn_final(float* __restrict__ out,
                                                 const float* __restrict__ x,
                                                 int n4)
{
    int i = blockIdx.x * 256 + threadIdx.x;
    if (i >= n4) return;
    float4 a = ((float4*)out)[i];
    float4 b = ((const float4*)x)[i];
    a.x = tanhf(a.x + b.x);
    a.y = tanhf(a.y + b.y);
    a.z = tanhf(a.z + b.z);
    a.w = tanhf(a.w + b.w);
    ((float4*)out)[i] = a;
}

// ---------------------------------------------------------------------------
extern "C" void kernel_launch(void* const* d_in, const int* in_sizes, int n_in,
                              void* d_out, int out_size, void* d_ws, size_t ws_size,
                              hipStream_t stream)
{
    const float* x    = (const float*)d_in[0];   // [1, N, 128]
    const float* Ws   = (const float*)d_in[1];   // [2, 2, 128, 128]
    const float* bs   = (const float*)d_in[2];   // [2, 2, 128]
    const float* vals = (const float*)d_in[3];   // [2, E]
    const int*   rows = (const int*)  d_in[4];   // [2, E]
    const int*   cols = (const int*)  d_in[5];   // [2, E]
    float* out = (float*)d_out;                  // [N, 128] = h_sum accumulator

    float* hA = (float*)d_ws;                    // 51.2 MB
    float* hB = hA + (size_t)NNODES * DIM;       // 51.2 MB

    const int n4       = (NNODES * DIM) / 4;
    const int zeroGrid = (n4 + 255) / 256;
    const int gemmGrid = (NNODES + BR - 1) / BR;
    const int spmmGrid = NEDGES / 256;           // 32 edges/wave * 8 waves/block

    gcn_zero<<<zeroGrid, 256, 0, stream>>>((float4*)out, n4);

    for (int s = 0; s < 2; ++s) {
        const float* W0 = Ws + ((size_t)(s * 2 + 0)) * DIM * DIM;
        const float* W1 = Ws + ((size_t)(s * 2 + 1)) * DIM * DIM;
        const float* b0 = bs + ((size_t)(s * 2 + 0)) * DIM;
        const float* b1 = bs + ((size_t)(s * 2 + 1)) * DIM;
        const float* ev = vals + (size_t)s * NEDGES;
        const int*   er = rows + (size_t)s * NEDGES;
        const int*   ec = cols + (size_t)s * NEDGES;

        // layer 0: hA = x @ W0^T + b0
        gcn_gemm<<<gemmGrid, 256, 0, stream>>>(x, W0, b0, hA, NNODES, 0);
        // hB = spmm(hA)
        gcn_zero<<<zeroGrid, 256, 0, stream>>>((float4*)hB, n4);
        gcn_spmm<<<spmmGrid, 256, 0, stream>>>(ev, er, ec, hA, hB, NEDGES);
        // layer 1: hA = tanh(hB) @ W1^T + b1   (tanh fused into A staging)
        gcn_gemm<<<gemmGrid, 256, 0, stream>>>(hB, W1, b1, hA, NNODES, 1);
        // h_sum += spmm(hA)   (accumulate straight into d_out)
        gcn_spmm<<<spmmGrid, 256, 0, stream>>>(ev, er, ec, hA, out, NEDGES);
    }

    // out = tanh(h_sum + x)
    gcn_final<<<zeroGrid, 256, 0, stream>>>(out, x, n4);
}